// GCN_17961553232569
// MI455X (gfx1250) — compile-verified
//
#include <hip/hip_runtime.h>
#include <hip/hip_bf16.h>

typedef __attribute__((ext_vector_type(2))) float v2f;
typedef __attribute__((ext_vector_type(8))) float v8f;

#define DIN 64
#define DH  64

// ---------------------------------------------------------------------------
// K0: init deg=1 (self loop), zero agg buffer, zero d_out
__global__ void k_init(float* __restrict__ deg, float* __restrict__ agg,
                       float* __restrict__ out, int N) {
    int tid = blockIdx.x * blockDim.x + threadIdx.x;   // over N*64
    if (tid < N * DH) agg[tid] = 0.0f;
    if (tid < N)      deg[tid] = 1.0f;
    if (tid < 128)    out[tid] = 0.0f;
}

// K1: in-degree via atomics over dst
__global__ void k_degree(const int* __restrict__ dst, float* __restrict__ deg, int E) {
    int e = blockIdx.x * blockDim.x + threadIdx.x;
    if (e < E) atomicAdd(&deg[dst[e]], 1.0f);
}

// K2: dinv = rsqrt(deg)   (deg >= 1 always, self-loop)
__global__ void k_dinv(const float* __restrict__ deg, float* __restrict__ dinv, int N) {
    int i = blockIdx.x * blockDim.x + threadIdx.x;
    if (i < N) dinv[i] = rsqrtf(deg[i]);
}

// ---------------------------------------------------------------------------
// WMMA GEMM:  H[N,64] = op(A[N,64]) @ W[64,64]  (+ rowvec[64] broadcast)
// op = relu if RELU_A.  One wave computes one 16x16 tile of H.
// blockDim = 128 (4 waves); wave w handles column tile w (64 cols / 16).
// V_WMMA_F32_16X16X4_F32 lane layout (ISA 7.12.2):
//   A: lane L: M = L&15, VGPR0 holds K = 2*(L>>4), VGPR1 holds K+1
//   B: lane L: N = L&15, VGPR0 holds K = 2*(L>>4), VGPR1 holds K+1
//   C/D: lane L: N = L&15, VGPRv holds M = 8*(L>>4) + v
template <bool RELU_A>
__global__ void k_gemm64(const float* __restrict__ A, const float* __restrict__ W,
                         const float* __restrict__ rowvec,   // nullptr -> zero init
                         float* __restrict__ H, int N) {
    const int lane  = threadIdx.x & 31;
    const int wave  = threadIdx.x >> 5;          // column tile 0..3
    const int m15   = lane & 15;
    const int khalf = (lane >> 4) << 1;          // 0 or 2
    const int row   = blockIdx.x * 16 + m15;     // A row for this lane
    const int col   = wave * 16 + m15;           // B/D column for this lane

    v8f c;
    const float cinit = rowvec ? rowvec[col] : 0.0f;
#pragma unroll
    for (int v = 0; v < 8; ++v) c[v] = cinit;

    const float* Arow = A + row * DIN;
#pragma unroll
    for (int kb = 0; kb < DIN; kb += 4) {
        v2f a, b;
        float a0 = Arow[kb + khalf];
        float a1 = Arow[kb + khalf + 1];
        if (RELU_A) { a0 = fmaxf(a0, 0.0f); a1 = fmaxf(a1, 0.0f); }
        a.x = a0; a.y = a1;
        b.x = W[(kb + khalf) * DH + col];
        b.y = W[(kb + khalf + 1) * DH + col];
        c = __builtin_amdgcn_wmma_f32_16x16x4_f32(false, a, false, b,
                                                  (short)0, c, false, false);
    }

    const int rbase = blockIdx.x * 16 + ((lane >> 4) << 3);
#pragma unroll
    for (int v = 0; v < 8; ++v)
        H[(rbase + v) * DH + col] = c[v];
}

// ---------------------------------------------------------------------------
// Scatter: agg[dst] += h[src] * dinv[src]*dinv[dst].  16 threads per edge,
// each thread one float4 group -> coalesced gather + 4 f32 atomics.
__global__ void k_scatter(const float* __restrict__ H, float* __restrict__ AGG,
                          const int* __restrict__ src, const int* __restrict__ dst,
                          const float* __restrict__ dinv, int E) {
    int tid = blockIdx.x * blockDim.x + threadIdx.x;   // over E*16
    if (tid >= E * 16) return;
    int e = tid >> 4;
    int g = (tid & 15) << 2;
    int s = src[e], d = dst[e];
    float w = dinv[s] * dinv[d];
    float4 hv = *(const float4*)(H + s * DH + g);
    float* o = AGG + d * DH + g;
    atomicAdd(o + 0, hv.x * w);
    atomicAdd(o + 1, hv.y * w);
    atomicAdd(o + 2, hv.z * w);
    atomicAdd(o + 3, hv.w * w);
}

// ---------------------------------------------------------------------------
// Finish conv1: C = AGG + H*dinv^2 + b ; re-zero AGG for the second scatter.
__global__ void k_finish1(const float* __restrict__ H, float* __restrict__ AGG,
                          const float* __restrict__ dinv, const float* __restrict__ b,
                          float* __restrict__ C, int N) {
    int tid = blockIdx.x * blockDim.x + threadIdx.x;   // over N*64
    if (tid >= N * DH) return;
    int i = tid >> 6, j = tid & 63;
    float di = dinv[i];
    C[tid] = AGG[tid] + H[tid] * di * di + b[j];
    AGG[tid] = 0.0f;
}

// ---------------------------------------------------------------------------
// Root epilogue: rowvec[j] = sum_k relu(x[root,k]) * W2[64+k, j]
//                out[j]    = c1[root, j]                       (single block, 64 thr)
__global__ void k_root(const float* __restrict__ x, const float* __restrict__ c1,
                       const float* __restrict__ W2, const int* __restrict__ rootp,
                       float* __restrict__ rowvec, float* __restrict__ out) {
    int j = threadIdx.x;                   // 0..63
    int root = rootp[0];
    float acc = 0.0f;
#pragma unroll 8
    for (int k = 0; k < DIN; ++k)
        acc += fmaxf(x[root * DIN + k], 0.0f) * W2[(DH + k) * 64 + j];
    rowvec[j] = acc;
    out[j] = c1[root * DH + j];
}

// ---------------------------------------------------------------------------
// Finish conv2 + column mean: acc_j += relu(AGG + H*dinv^2 + b2)[r,j] over rows,
// one 64-thread block per 64 rows (contiguous row reads), 64 atomics/block.
__global__ void k_finish2(const float* __restrict__ H, const float* __restrict__ AGG,
                          const float* __restrict__ dinv, const float* __restrict__ b,
                          float* __restrict__ out, int N, float invN) {
    int j = threadIdx.x;                   // column 0..63
    int r0 = blockIdx.x * 64;
    int r1 = min(r0 + 64, N);
    float bj = b[j], acc = 0.0f;
    for (int r = r0; r < r1; ++r) {
        float di = dinv[r];
        float v = AGG[r * DH + j] + H[r * DH + j] * di * di + bj;
        acc += fmaxf(v, 0.0f);
    }
    atomicAdd(&out[64 + j], acc * invN);
}

// ---------------------------------------------------------------------------
extern "C" void kernel_launch(void* const* d_in, const int* in_sizes, int n_in,
                              void* d_out, int out_size, void* d_ws, size_t ws_size,
                              hipStream_t stream) {
    const float* x     = (const float*)d_in[0];
    const int*   ei    = (const int*)d_in[1];
    const int*   rootp = (const int*)d_in[2];
    const float* W1    = (const float*)d_in[3];
    const float* b1    = (const float*)d_in[4];
    const float* W2    = (const float*)d_in[5];
    const float* b2    = (const float*)d_in[6];
    float* out = (float*)d_out;

    const int N = in_sizes[0] / DIN;       // 50000 (divisible by 16)
    const int E = in_sizes[1] / 2;
    const int* src = ei;
    const int* dst = ei + E;

    // workspace layout (floats)
    float* ws   = (float*)d_ws;
    float* deg  = ws;                      // N
    float* dinv = deg + N;                 // N
    float* h    = dinv + N;                // N*64  (h1, then reused as h2)
    float* agg  = h + N * DH;              // N*64  (agg1, then reused as agg2)
    float* c1   = agg + N * DH;            // N*64
    float* rvec = c1 + N * DH;             // 64

    const int T = 256;
    const int nNF = N * DH;

    // K0: init
    k_init<<<(nNF + T - 1) / T, T, 0, stream>>>(deg, agg, out, N);
    // K1: degree
    k_degree<<<(E + T - 1) / T, T, 0, stream>>>(dst, deg, E);
    // K2: dinv
    k_dinv<<<(N + T - 1) / T, T, 0, stream>>>(deg, dinv, N);
    // K3: h1 = x @ W1  (WMMA f32)
    k_gemm64<false><<<N / 16, 128, 0, stream>>>(x, W1, nullptr, h, N);
    // K4: scatter edges into agg1
    k_scatter<<<(E * 16 + T - 1) / T, T, 0, stream>>>(h, agg, src, dst, dinv, E);
    // K5: c1 = agg1 + selfloop + b1 ; zero agg for round 2
    k_finish1<<<(nNF + T - 1) / T, T, 0, stream>>>(h, agg, dinv, b1, c1, N);
    // K6: root row-vector + out[0:64] = c1[root]
    k_root<<<1, 64, 0, stream>>>(x, c1, W2, rootp, rvec, out);
    // K7: h2 = relu(c1) @ W2[:64] + rvec  (WMMA f32)
    k_gemm64<true><<<N / 16, 128, 0, stream>>>(c1, W2, rvec, h, N);
    // K8: scatter edges into agg2
    k_scatter<<<(E * 16 + T - 1) / T, T, 0, stream>>>(h, agg, src, dst, dinv, E);
    // K9: c2 = relu(agg2 + selfloop + b2); out[64:128] = column mean
    k_finish2<<<(N + 63) / 64, 64, 0, stream>>>(h, agg, dinv, b2, out, N, 1.0f / (float)N);
}